// Seq2Seq_8340826489455
// MI455X (gfx1250) — compile-verified
//
#include <hip/hip_runtime.h>
#include <math.h>

// ---------------------------------------------------------------------------
// Seq2Seq decoder only (encoder state is discarded by the reference = dead
// code). Per step t = 1..47 (serial chain; next token = argmax of step t):
//   x  = dec_emb[cur_tok]                           (32 x 512)
//   l0/l1: gates = x @ Wih[l].T + bih[l] + bhh[l]   (h0 = c0 = 0 each step)
//          x = sigmoid(o) * tanh( sigmoid(i) * tanh(g) )
//   logits_t = x @ fc_W.T + fc_b -> d_out[t]        (32 x 32000)
//   cur_tok  = tf_mask[t] ? trg[t] : argmax(logits_t)
//
// v_wmma_f32_16x16x4_f32 keeps f32-exact math (GEMMs are L2/latency bound at
// batch 32). Both decoder layers are fused in ONE 1024-thread workgroup with
// h0 in LDS; the FC GEMM (2000 waves) streams fc_W from the 192MB L2.
// All fragment loads use uniform-base + 32-bit lane offset so the backend
// emits saddr+scale_offset addressing, no 64-bit VALU adds in the loop.
// ---------------------------------------------------------------------------

typedef float v2f __attribute__((ext_vector_type(2)));
typedef float v8f __attribute__((ext_vector_type(8)));

#define VOCAB   32000
#define EMB     512
#define HID     512
#define BATCH   32
#define TRG_LEN 48

__device__ __forceinline__ v8f wmma4f32(v2f a, v2f b, v8f c) {
    return __builtin_amdgcn_wmma_f32_16x16x4_f32(
        false, a, false, b, (short)0, c, false, false);
}

// Branch-free activations on the hardware transcendental pipe (v_exp_f32).
__device__ __forceinline__ float sig_(float x)  { return 1.0f / (1.0f + __expf(-x)); }
__device__ __forceinline__ float tanh_(float x) { return 1.0f - 2.0f / (__expf(2.0f * x) + 1.0f); }

// ---------------------------------------------------------------------------
// init: cur_tok = trg[0], zero output row 0 (grid covers 32*32000 threads)
// ---------------------------------------------------------------------------
__global__ __launch_bounds__(256) void k_init(const int* __restrict__ trg,
                                              int* __restrict__ cur_tok,
                                              float* __restrict__ out0) {
    int i = blockIdx.x * 256 + threadIdx.x;
    if (i < BATCH) cur_tok[i] = trg[i];           // trg[0][b]
    if (i < BATCH * VOCAB) out0[i] = 0.0f;        // outputs[0] stays zero
}

// ---------------------------------------------------------------------------
// Fused 2-layer decoder step: one workgroup, 32 waves (1024 thr) on one WGP.
// Layer 0 gathers A rows straight from dec_emb[tok]; h0 lives in LDS (64KB);
// layer 1 reads A fragments from LDS (ds_load_b64) and writes h1 to global.
// Each wave owns one 16-wide column tile; i/f/g/o reduce to i/g/o (c0=h0=0).
// ---------------------------------------------------------------------------
__global__ __launch_bounds__(1024) void k_dec(const float* __restrict__ emb,  // (32000, 512)
                                              const int*   __restrict__ tok,  // 32
                                              const float* __restrict__ Wih,  // (2, 2048, 512)
                                              const float* __restrict__ bih,  // (2, 2048)
                                              const float* __restrict__ bhh,  // (2, 2048)
                                              float* __restrict__ Hout) {     // 32x512
    __shared__ float h0buf[BATCH * HID];          // 64 KB of the 320 KB WGP LDS

    const int lane  = threadIdx.x & 31;
    const int wave  = threadIdx.x >> 5;           // 0..31 -> column tile
    const int mlo   = lane & 15;
    const int khalf = (lane >> 4) * 2;            // K sub-pair per half-wave
    const int n     = wave * 16 + mlo;            // this lane's output column
    const int mbase = 8 * (lane >> 4);            // C/D row base: j + 8*(lane/16)

    // ---------------- layer 0: A = dec_emb[tok] ----------------
    const int aoff0 = tok[mlo]      * EMB + khalf;   // 32-bit element offsets
    const int aoff1 = tok[mlo + 16] * EMB + khalf;
    const int wi = (n)        * HID + khalf;
    const int wg = (1024 + n) * HID + khalf;
    const int wo = (1536 + n) * HID + khalf;

    v8f ci0 = {}, ci1 = {}, cg0 = {}, cg1 = {}, co0 = {}, co1 = {};
#pragma unroll 8
    for (int k = 0; k < HID; k += 4) {
        v2f a0 = *(const v2f*)(emb + aoff0 + k);
        v2f a1 = *(const v2f*)(emb + aoff1 + k);
        v2f bi = *(const v2f*)(Wih + wi + k);
        v2f bg = *(const v2f*)(Wih + wg + k);
        v2f bo = *(const v2f*)(Wih + wo + k);
        ci0 = wmma4f32(a0, bi, ci0);  ci1 = wmma4f32(a1, bi, ci1);
        cg0 = wmma4f32(a0, bg, cg0);  cg1 = wmma4f32(a1, bg, cg1);
        co0 = wmma4f32(a0, bo, co0);  co1 = wmma4f32(a1, bo, co1);
    }
    {
        float bi_s = bih[n]        + bhh[n];
        float bg_s = bih[1024 + n] + bhh[1024 + n];
        float bo_s = bih[1536 + n] + bhh[1536 + n];
#pragma unroll
        for (int j = 0; j < 8; ++j) {
            int m0 = j + mbase;
            float cv0 = sig_(ci0[j] + bi_s) * tanh_(cg0[j] + bg_s);
            h0buf[m0 * HID + n] = sig_(co0[j] + bo_s) * tanh_(cv0);
            float cv1 = sig_(ci1[j] + bi_s) * tanh_(cg1[j] + bg_s);
            h0buf[(m0 + 16) * HID + n] = sig_(co1[j] + bo_s) * tanh_(cv1);
        }
    }
    __syncthreads();

    // ---------------- layer 1: A = h0 (LDS) ----------------
    const float* W1  = Wih + 4 * HID * HID;       // layer-1 Wih
    const int la0 = mlo        * HID + khalf;     // LDS element offsets
    const int la1 = (mlo + 16) * HID + khalf;

    ci0 = (v8f){}; ci1 = (v8f){}; cg0 = (v8f){};
    cg1 = (v8f){}; co0 = (v8f){}; co1 = (v8f){};
#pragma unroll 8
    for (int k = 0; k < HID; k += 4) {
        v2f a0 = *(const v2f*)(&h0buf[la0 + k]);  // ds_load_b64
        v2f a1 = *(const v2f*)(&h0buf[la1 + k]);
        v2f bi = *(const v2f*)(W1 + wi + k);
        v2f bg = *(const v2f*)(W1 + wg + k);
        v2f bo = *(const v2f*)(W1 + wo + k);
        ci0 = wmma4f32(a0, bi, ci0);  ci1 = wmma4f32(a1, bi, ci1);
        cg0 = wmma4f32(a0, bg, cg0);  cg1 = wmma4f32(a1, bg, cg1);
        co0 = wmma4f32(a0, bo, co0);  co1 = wmma4f32(a1, bo, co1);
    }
    {
        float bi_s = bih[2048 + n]        + bhh[2048 + n];
        float bg_s = bih[2048 + 1024 + n] + bhh[2048 + 1024 + n];
        float bo_s = bih[2048 + 1536 + n] + bhh[2048 + 1536 + n];
#pragma unroll
        for (int j = 0; j < 8; ++j) {
            int m0 = j + mbase;
            float cv0 = sig_(ci0[j] + bi_s) * tanh_(cg0[j] + bg_s);
            Hout[m0 * HID + n] = sig_(co0[j] + bo_s) * tanh_(cv0);
            float cv1 = sig_(ci1[j] + bi_s) * tanh_(cg1[j] + bg_s);
            Hout[(m0 + 16) * HID + n] = sig_(co1[j] + bo_s) * tanh_(cv1);
        }
    }
}

// ---------------------------------------------------------------------------
// FC: logits(32 x 32000) = H @ fc_W.T + fc_b -> d_out row t.
// 2000 waves; each owns a 32x16 tile (two 16x16 f32 WMMA accumulators).
// Fully unrolled K: all loads become saddr + lane-offset + immediate.
// ---------------------------------------------------------------------------
__global__ __launch_bounds__(256) void k_fc(const float* __restrict__ H,    // 32x512
                                            const float* __restrict__ fcW,  // 32000x512
                                            const float* __restrict__ fcb,  // 32000
                                            float* __restrict__ outRow) {   // 32x32000
    int wave = (blockIdx.x * 256 + threadIdx.x) >> 5;   // 0..1999
    int lane = threadIdx.x & 31;
    if (wave >= VOCAB / 16) return;

    const int nl    = lane & 15;
    const int khalf = (lane >> 4) * 2;
    const int v     = wave * 16 + nl;                   // vocab column

    const int aoff0 = nl        * HID + khalf;
    const int aoff1 = (nl + 16) * HID + khalf;
    const int boff  = v * HID + khalf;                  // < 16.4M, fits int

    v8f c0 = {}, c1 = {};
#pragma unroll
    for (int k = 0; k < HID; k += 4) {
        v2f a0 = *(const v2f*)(H + aoff0 + k);
        v2f a1 = *(const v2f*)(H + aoff1 + k);
        v2f b  = *(const v2f*)(fcW + boff + k);
        c0 = wmma4f32(a0, b, c0);
        c1 = wmma4f32(a1, b, c1);
    }

    float bias  = fcb[v];
    int   mbase = 8 * (lane >> 4);
#pragma unroll
    for (int j = 0; j < 8; ++j) {
        int m = j + mbase;
        outRow[m * VOCAB + v]        = c0[j] + bias;    // < 1.03M elems: int ok
        outRow[(m + 16) * VOCAB + v] = c1[j] + bias;
    }
}

// ---------------------------------------------------------------------------
// Deterministic first-max argmax per batch row + teacher-forcing select.
// ---------------------------------------------------------------------------
__global__ __launch_bounds__(256) void k_argmax(const float* __restrict__ logitsRow,
                                                const int* __restrict__ trg,
                                                const unsigned char* __restrict__ tf,
                                                int t,
                                                int* __restrict__ cur_tok) {
    __shared__ float sv[256];
    __shared__ int   si[256];
    int b   = blockIdx.x;
    int tid = threadIdx.x;
    const float* row = logitsRow + (size_t)b * VOCAB;

    float best = -__builtin_huge_valf();
    int   bidx = 0x7fffffff;
    for (int v = tid; v < VOCAB; v += 256) {            // 125 iters exactly
        float val = row[v];
        if (val > best) { best = val; bidx = v; }       // strict >: first max
    }
    sv[tid] = best; si[tid] = bidx;
    __syncthreads();
    for (int s = 128; s > 0; s >>= 1) {
        if (tid < s) {
            float v2 = sv[tid + s]; int i2 = si[tid + s];
            if (v2 > sv[tid] || (v2 == sv[tid] && i2 < si[tid])) { sv[tid] = v2; si[tid] = i2; }
        }
        __syncthreads();
    }
    if (tid == 0)
        cur_tok[b] = tf[t] ? trg[(size_t)t * BATCH + b] : si[0];
}

// ---------------------------------------------------------------------------
extern "C" void kernel_launch(void* const* d_in, const int* in_sizes, int n_in,
                              void* d_out, int out_size, void* d_ws, size_t ws_size,
                              hipStream_t stream) {
    (void)in_sizes; (void)n_in; (void)out_size; (void)ws_size;
    const int*           trg     = (const int*)d_in[1];
    const unsigned char* tf      = (const unsigned char*)d_in[2];   // jax bool: 1 byte
    const float*         dec_emb = (const float*)d_in[8];
    const float*         dec_Wih = (const float*)d_in[9];           // (2, 2048, 512)
    const float*         dec_bih = (const float*)d_in[11];          // (2, 2048)
    const float*         dec_bhh = (const float*)d_in[12];
    const float*         fc_W    = (const float*)d_in[13];          // (32000, 512)
    const float*         fc_b    = (const float*)d_in[14];
    float*               out     = (float*)d_out;                   // (48, 32, 32000)

    char* ws      = (char*)d_ws;
    int*  cur_tok = (int*)ws;                                       // 32 ints
    float* h1     = (float*)(ws + 256);                             // 32x512

    k_init<<<(BATCH * VOCAB + 255) / 256, 256, 0, stream>>>(trg, cur_tok, out);

    const size_t stepElems = (size_t)BATCH * VOCAB;
    for (int t = 1; t < TRG_LEN; ++t) {
        // fused embedding + 2 LSTM layers: one WGP, 32 waves, h0 in LDS
        k_dec<<<1, 1024, 0, stream>>>(dec_emb, cur_tok,
                                      dec_Wih, dec_bih, dec_bhh, h1);
        // FC -> logits row t (2000 waves; fc_W hot in L2)
        k_fc<<<250, 256, 0, stream>>>(h1, fc_W, fc_b, out + (size_t)t * stepElems);
        // next token (deterministic argmax + teacher forcing)
        k_argmax<<<BATCH, 256, 0, stream>>>(out + (size_t)t * stepElems,
                                            trg, tf, t, cur_tok);
    }
}